// GCN_50208167690349
// MI455X (gfx1250) — compile-verified
//
#include <hip/hip_runtime.h>
#include <hip/hip_bf16.h>

typedef __attribute__((ext_vector_type(2))) float v2f;
typedef __attribute__((ext_vector_type(8))) float v8f;

#define C128 128

// ---------------------------------------------------------------------------
// Zero the aggregation accumulator (d_out) and the degree array.
// ---------------------------------------------------------------------------
__global__ __launch_bounds__(256) void gcn_zero_kernel(float* __restrict__ agg,
                                                       float* __restrict__ deg,
                                                       long long n_agg, int N) {
  long long i = (long long)blockIdx.x * blockDim.x + threadIdx.x;
  if (i < n_agg) agg[i] = 0.0f;
  if (i < N)     deg[i] = 0.0f;
}

// ---------------------------------------------------------------------------
// deg[dst[e]] += 1 over all E edges (self-loops handled analytically later).
// ---------------------------------------------------------------------------
__global__ __launch_bounds__(256) void gcn_deg_kernel(const long long* __restrict__ dst,
                                                      float* __restrict__ deg, int E) {
  int e = blockIdx.x * blockDim.x + threadIdx.x;
  if (e < E) {
    atomicAdd(&deg[(int)dst[e]], 1.0f);
  }
}

// ---------------------------------------------------------------------------
// dinv[i] = rsqrt(deg[i] + 1)   (the +1 is the self-loop; deg+1 > 0 always)
// ---------------------------------------------------------------------------
__global__ __launch_bounds__(256) void gcn_dinv_kernel(const float* __restrict__ deg,
                                                       float* __restrict__ dinv, int N) {
  int i = blockIdx.x * blockDim.x + threadIdx.x;
  if (i < N) {
    dinv[i] = rsqrtf(deg[i] + 1.0f);
  }
}

// ---------------------------------------------------------------------------
// h = x @ W  via V_WMMA_F32_16X16X4_F32.
// Block = 256 threads = 8 waves. Wave w computes the 16x16 tile
// rows [16*blockIdx.x, +16) x cols [16*w, +16). K loop: 32 steps of K=4.
//
// fp32 WMMA lane layout (ISA 7.12.2):
//   A (16x4): lanes 0-15 -> M=lane, V0=K0,V1=K1 ; lanes 16-31 -> V0=K2,V1=K3
//   B (4x16): lanes 0-15 -> N=lane, V0=K0,V1=K1 ; lanes 16-31 -> V0=K2,V1=K3
//   C/D (16x16): VGPR r -> M=r (lanes 0-15) / M=8+r (lanes 16-31), N=lane&15
// ---------------------------------------------------------------------------
__global__ __launch_bounds__(256) void gcn_gemm_kernel(const float* __restrict__ x,
                                                       const float* __restrict__ W,
                                                       float* __restrict__ h, int N) {
  const int wave = threadIdx.x >> 5;
  const int lane = threadIdx.x & 31;
  const int half = lane >> 4;      // 0: K pair {0,1}, 1: K pair {2,3}
  const int l16  = lane & 15;
  const int row0 = blockIdx.x * 16;
  const int col0 = wave * 16;

  // Clamp A-row reads for a possible ragged last block (stores are guarded).
  const int arow = (row0 + l16 < N) ? (row0 + l16) : (N - 1);
  const float* __restrict__ xrow = x + (size_t)arow * C128;
  const float* __restrict__ wcol = W + col0 + l16;

  v8f c = {};
#pragma unroll
  for (int k = 0; k < C128; k += 4) {
    const int ka = k + 2 * half;
    v2f a, b;
    a.x = xrow[ka];
    a.y = xrow[ka + 1];
    b.x = wcol[(size_t)ka * C128];
    b.y = wcol[(size_t)(ka + 1) * C128];
    // D = A*B + C  (8-arg form: neg_a, A, neg_b, B, c_mod, C, reuse_a, reuse_b)
    c = __builtin_amdgcn_wmma_f32_16x16x4_f32(false, a, false, b, (short)0, c,
                                              false, false);
  }

  const int mbase = row0 + half * 8;
#pragma unroll
  for (int r = 0; r < 8; ++r) {
    const int m = mbase + r;
    if (m < N) h[(size_t)m * C128 + col0 + l16] = c[r];
  }
}

// ---------------------------------------------------------------------------
// Edge aggregation: one wave per edge, each lane owns 4 channels.
//   agg[dst] += h[src] * (dinv[src]*dinv[dst])
// h and agg both live in the 192MB L2, so the gather + atomic RMW traffic
// stays on-chip.
// ---------------------------------------------------------------------------
__global__ __launch_bounds__(256) void gcn_agg_kernel(const long long* __restrict__ src,
                                                      const long long* __restrict__ dst,
                                                      const float* __restrict__ h,
                                                      const float* __restrict__ dinv,
                                                      float* __restrict__ agg, int E) {
  const int wave = threadIdx.x >> 5;
  const int lane = threadIdx.x & 31;
  const long long e = (long long)blockIdx.x * 8 + wave;
  if (e >= E) return;

  const int s = (int)src[e];
  const int d = (int)dst[e];
  const float nrm = dinv[s] * dinv[d];

  const float4 hv = *reinterpret_cast<const float4*>(h + (size_t)s * C128 + lane * 4);
  float* ap = agg + (size_t)d * C128 + lane * 4;
  atomicAdd(ap + 0, hv.x * nrm);
  atomicAdd(ap + 1, hv.y * nrm);
  atomicAdd(ap + 2, hv.z * nrm);
  atomicAdd(ap + 3, hv.w * nrm);
}

// ---------------------------------------------------------------------------
// Epilogue: out = relu(agg + h*dinv^2 (self-loop) + b), in place on d_out.
// ---------------------------------------------------------------------------
__global__ __launch_bounds__(256) void gcn_final_kernel(float* __restrict__ out,
                                                        const float* __restrict__ h,
                                                        const float* __restrict__ dinv,
                                                        const float* __restrict__ bias,
                                                        int N) {
  const long long idx = (long long)blockIdx.x * blockDim.x + threadIdx.x;
  const long long total = (long long)N * C128;
  if (idx >= total) return;
  const int i = (int)(idx >> 7);
  const int cch = (int)(idx & (C128 - 1));
  const float di = dinv[i];
  const float v = out[idx] + h[idx] * di * di + bias[cch];
  out[idx] = v > 0.0f ? v : 0.0f;
}

// ---------------------------------------------------------------------------
extern "C" void kernel_launch(void* const* d_in, const int* in_sizes, int n_in,
                              void* d_out, int out_size, void* d_ws, size_t ws_size,
                              hipStream_t stream) {
  const int N = in_sizes[0] / C128;       // 100000
  const int E = in_sizes[1] / 2;          // 1600000

  const float*     x   = (const float*)d_in[0];
  const long long* ei  = (const long long*)d_in[1];   // int64 [2, E]
  const float*     W   = (const float*)d_in[2];
  const float*     b   = (const float*)d_in[3];
  const long long* src = ei;        // edge_index[0]
  const long long* dst = ei + E;    // edge_index[1]

  float* out  = (float*)d_out;                  // doubles as agg accumulator
  float* h    = (float*)d_ws;                   // [N, 128]
  float* deg  = h + (size_t)N * C128;           // [N]
  float* dinv = deg + N;                        // [N]

  const long long total = (long long)N * C128;
  const int T = 256;

  gcn_zero_kernel<<<(unsigned)((total + T - 1) / T), T, 0, stream>>>(out, deg, total, N);
  gcn_gemm_kernel<<<(unsigned)((N + 15) / 16), T, 0, stream>>>(x, W, h, N);
  gcn_deg_kernel<<<(unsigned)((E + T - 1) / T), T, 0, stream>>>(dst, deg, E);
  gcn_dinv_kernel<<<(unsigned)((N + T - 1) / T), T, 0, stream>>>(deg, dinv, N);
  gcn_agg_kernel<<<(unsigned)((E + 7) / 8), T, 0, stream>>>(src, dst, h, dinv, out, E);
  gcn_final_kernel<<<(unsigned)((total + T - 1) / T), T, 0, stream>>>(out, h, dinv, b, N);
}